// PrimaryCapsuleLayer_29557964931628
// MI455X (gfx1250) — compile-verified
//
#include <hip/hip_runtime.h>

typedef __attribute__((ext_vector_type(2))) float v2f;
typedef __attribute__((ext_vector_type(4))) float v4f;
typedef __attribute__((ext_vector_type(8))) float v8f;

#define BATCH       131072
#define SCALAR_FEAT 256
#define NCOLS       256        // OUT_CAPS * (CAPS_DIM/2)
#define OUT_STRIDE  512        // OUT_CAPS * CAPS_DIM floats per batch row
#define LDS_STRIDE  260        // padded row stride (floats): 260 % 64 == 4 -> no bank conflicts

// ---------------------------------------------------------------------------
// GEMM + capsule interleave:  out[b, cap*16 + j]   = (Xs @ Ws + bs)[b, cap*8+j]
//                             out[b, cap*16 + 8+j] = 0
// One block per 16-row M tile. 8 waves; wave w handles columns [32w, 32w+32).
// Epilogue round-trips accumulators through LDS to emit coalesced b128 NT stores.
// ---------------------------------------------------------------------------
__global__ __launch_bounds__(256) void capsule_gemm_kernel(
    const float* __restrict__ Xs,   // [BATCH, 256]
    const float* __restrict__ Ws,   // [256, 256]
    const float* __restrict__ bs,   // [256]
    float* __restrict__ out)        // [BATCH, 32, 16]
{
    __shared__ float smem[16 * LDS_STRIDE];   // 16.25 KB; A tile, then C tile

    const int tid  = threadIdx.x;
    const int lane = tid & 31;
    const int wave = tid >> 5;
    const int m0   = blockIdx.x * 16;

    // ---- Stage A tile (16 x 256 f32) into LDS with padded rows, coalesced ----
    {
        const v4f* __restrict__ src = (const v4f*)(Xs + (size_t)m0 * SCALAR_FEAT);
#pragma unroll
        for (int i = 0; i < 4; ++i) {
            const int f   = i * 256 + tid;   // float4 index 0..1023
            const int row = f >> 6;          // 64 float4 per source row
            const int c4  = f & 63;
            *(v4f*)(smem + row * LDS_STRIDE + c4 * 4) = src[f];
        }
    }
    __syncthreads();

    const int n     = lane & 15;        // column within 16-wide tile
    const int khalf = (lane >> 4) * 2;  // lanes 0-15 -> K+0/+1, lanes 16-31 -> K+2/+3
    const int c0    = wave * 32;
    const int c1    = c0 + 16;

    // Bias folded into the WMMA C operand (bias depends on column only).
    const float bias0 = bs[c0 + n];
    const float bias1 = bs[c1 + n];
    v8f acc0, acc1;
#pragma unroll
    for (int i = 0; i < 8; ++i) { acc0[i] = bias0; acc1[i] = bias1; }

    const float* __restrict__ lds_row = smem + (size_t)n * LDS_STRIDE;

#pragma unroll 4
    for (int kk = 0; kk < SCALAR_FEAT; kk += 4) {
        const int kb = kk + khalf;

        // A fragment (16x4 f32): banks 4n+kb / 4n+kb+2 -> conflict-free
        v2f a = *(const v2f*)(lds_row + kb);

        // B fragments (4x16 f32) straight from L2-resident Ws
        v2f b0, b1;
        b0.x = Ws[(size_t)kb * NCOLS + c0 + n];
        b0.y = Ws[(size_t)(kb + 1) * NCOLS + c0 + n];
        b1.x = Ws[(size_t)kb * NCOLS + c1 + n];
        b1.y = Ws[(size_t)(kb + 1) * NCOLS + c1 + n];

        acc0 = __builtin_amdgcn_wmma_f32_16x16x4_f32(false, a, false, b0,
                                                     (short)0, acc0, false, false);
        acc1 = __builtin_amdgcn_wmma_f32_16x16x4_f32(false, a, false, b1,
                                                     (short)0, acc1, false, false);
    }

    // ---- Epilogue: acc -> LDS (conflict-free scatter), then coalesced NT b128 ----
    __syncthreads();   // everyone done reading the A tile
#pragma unroll
    for (int i = 0; i < 8; ++i) {
        const int row = i + (lane >> 4) * 8;   // C/D VGPR i holds row i / i+8
        smem[row * LDS_STRIDE + c0 + n] = acc0[i];
        smem[row * LDS_STRIDE + c1 + n] = acc1[i];
    }
    __syncthreads();

    // Block's output region is 16 rows x 512 floats = 2048 contiguous float4s.
    // Row layout per capsule: [8 gemm floats][8 zeros] = quads {g,g,0,0}.
    v4f* __restrict__ obase = (v4f*)(out + (size_t)m0 * OUT_STRIDE);
#pragma unroll
    for (int i = 0; i < 8; ++i) {
        const int f   = i * 256 + tid;   // out float4 index 0..2047 (contiguous per store)
        const int row = f >> 7;          // 128 float4 per output row
        const int q   = f & 127;
        const int cap = q >> 2;
        const int sub = q & 3;
        v4f val = {0.0f, 0.0f, 0.0f, 0.0f};
        if (sub < 2)                     // scalar half; sub 2,3 are the zero half
            val = *(const v4f*)(smem + row * LDS_STRIDE + cap * 8 + sub * 4);
        __builtin_nontemporal_store(val, obase + f);
    }
}

// ---------------------------------------------------------------------------
// Streaming passthrough of x_vector (second tuple output), 128-bit NT copies.
// 6144 blocks * 256 threads * 8 float4 == 12582912 == BATCH*128*3/4 exactly.
// ---------------------------------------------------------------------------
__global__ __launch_bounds__(256) void vec_copy_kernel(
    const v4f* __restrict__ src, v4f* __restrict__ dst)
{
    const size_t base = (size_t)blockIdx.x * 2048 + threadIdx.x;
#pragma unroll
    for (int i = 0; i < 8; ++i) {
        const size_t idx = base + (size_t)i * 256;
        __builtin_nontemporal_store(__builtin_nontemporal_load(src + idx), dst + idx);
    }
}

extern "C" void kernel_launch(void* const* d_in, const int* in_sizes, int n_in,
                              void* d_out, int out_size, void* d_ws, size_t ws_size,
                              hipStream_t stream) {
    const float* Xs = (const float*)d_in[0];   // x_scalar [131072, 256]
    const float* Xv = (const float*)d_in[1];   // x_vector [131072, 128, 3]
    const float* Ws = (const float*)d_in[2];   // W_s [256, 256]
    const float* bs = (const float*)d_in[3];   // b_s [256]
    // d_in[4] (W_v), d_in[5] (b_v) are dead: their branch feeds an e3nn tensor
    // product with zero valid Clebsch-Gordan paths (output is identically 0).
    float* out = (float*)d_out;

    capsule_gemm_kernel<<<BATCH / 16, 256, 0, stream>>>(Xs, Ws, bs, out);

    float* vec_out = out + (size_t)BATCH * OUT_STRIDE;   // offset 67108864
    vec_copy_kernel<<<6144, 256, 0, stream>>>((const v4f*)Xv, (v4f*)vec_out);
}